// SparseConvolutionDownsample_65223373357587
// MI455X (gfx1250) — compile-verified
//
#include <hip/hip_runtime.h>
#include <hip/hip_bf16.h>

typedef __attribute__((ext_vector_type(2)))  float  v2f;
typedef __attribute__((ext_vector_type(8)))  float  v8f;

#define CIN  32
#define COUT 64
#define KEXP 128   // 4 kernel offsets * 32 Cin, expanded-K GEMM
#define WAVES_PER_BLOCK 4
#define ROWS_PER_WAVE 32   // two 16-row A tiles share each B fragment
#define BN_EPS 1e-4f
#define LEAK   0.333f

// Feature probing is only meaningful on the device pass; the host pass
// reports false for __has_builtin on amdgcn builtins even though calls
// inside device bodies type-check there.
#if defined(__HIP_DEVICE_COMPILE__)
#if !__has_builtin(__builtin_amdgcn_wmma_f32_16x16x4_f32)
#error "expected f32 WMMA builtin on gfx1250 (confirmed in round 1)"
#endif
#if __has_builtin(__builtin_amdgcn_global_load_async_to_lds_b32) && \
    __has_builtin(__builtin_amdgcn_s_wait_asynccnt)
#define HAVE_ASYNC_LDS 1
#endif
#endif

#ifdef HAVE_ASYNC_LDS
typedef __attribute__((address_space(1))) int* gptr_i32;
typedef __attribute__((address_space(3))) int* lptr_i32;
#endif

// ---------------------------------------------------------------- zero
__global__ void zero_kernel(float* __restrict__ out, int out_n,
                            float* __restrict__ stats) {
    const int i = blockIdx.x * blockDim.x + threadIdx.x;
    const int stride = gridDim.x * blockDim.x;
    for (int j = i; j < out_n; j += stride) out[j] = 0.f;
    if (i < 128) stats[i] = 0.f;
}

// ------------------------------------------------- gather-WMMA-scatter
// One wave handles 32 consecutive input rows (two 16-row M tiles).
// A (2 x 16x128) = expanded features, B (128x64) = all 4 weight slices.
__global__ __launch_bounds__(32 * WAVES_PER_BLOCK)
void conv_wmma_kernel(const float* __restrict__ feats,
                      const float* __restrict__ weight,   // [4][32][64] row-major
                      const int*   __restrict__ out_idx,
                      const int*   __restrict__ kern_idx,
                      float*       __restrict__ acc,      // [n_out][64]
                      int n_active) {
    // Stage weight transposed: Wt[n][kk] = weight[kk*64+n]  (32 KB LDS)
    __shared__ float Wt[COUT][KEXP];
    const int tid = threadIdx.x;
    for (int e = tid; e < KEXP * COUT; e += blockDim.x) {
        const int kk = e >> 6;
        const int n  = e & 63;
#ifdef HAVE_ASYNC_LDS
        // async DMA straight into LDS: transpose happens via per-lane LDS addr
        __builtin_amdgcn_global_load_async_to_lds_b32(
            (gptr_i32)(weight + e), (lptr_i32)&Wt[n][kk], 0, 0);
#else
        Wt[n][kk] = weight[e];
#endif
    }
#ifdef HAVE_ASYNC_LDS
    __builtin_amdgcn_s_wait_asynccnt(0);
#endif
    __syncthreads();

    const int lane = tid & 31;
    const int wave = tid >> 5;
    const int g    = lane >> 4;   // lane half (selects K pair / M half)
    const int ml   = lane & 15;
    const int tile = blockIdx.x * WAVES_PER_BLOCK + wave;
    const int row_base = tile * ROWS_PER_WAVE;
    if (row_base >= n_active) return;   // wave-uniform early out

    const int  r0  = row_base + ml;          // M tile 0
    const int  r1  = row_base + 16 + ml;     // M tile 1
    const bool ok0 = r0 < n_active;
    const bool ok1 = r1 < n_active;
    const int  kid0 = ok0 ? kern_idx[r0] : -1;
    const int  kid1 = ok1 ? kern_idx[r1] : -1;

    // A-frag feed: lane needs cols c = 4t + 2g + {0,1}, t=0..7
    float2 rp0[8], rp1[8];
#pragma unroll
    for (int t = 0; t < 8; ++t) {
        rp0[t] = make_float2(0.f, 0.f);
        rp1[t] = make_float2(0.f, 0.f);
    }
    if (ok0) {
        const float* f = feats + (size_t)r0 * CIN + 2 * g;
#pragma unroll
        for (int t = 0; t < 8; ++t) rp0[t] = *(const float2*)(f + 4 * t);
    }
    if (ok1) {
        const float* f = feats + (size_t)r1 * CIN + 2 * g;
#pragma unroll
        for (int t = 0; t < 8; ++t) rp1[t] = *(const float2*)(f + 4 * t);
    }

    v8f cacc[2][4] = {};   // 2 M-tiles x 4 N-tiles of 16x16 f32

    for (int w = 0; w < 4; ++w) {            // 32-wide K window = kernel offset
        const bool act0 = (w == kid0);
        const bool act1 = (w == kid1);
#pragma unroll
        for (int t = 0; t < 8; ++t) {        // K = 4*(8w+t) .. +3
            const int ks = 8 * w + t;
            v2f a0, a1;
            a0.x = act0 ? rp0[t].x : 0.f;
            a0.y = act0 ? rp0[t].y : 0.f;
            a1.x = act1 ? rp1[t].x : 0.f;
            a1.y = act1 ? rp1[t].y : 0.f;
#pragma unroll
            for (int n = 0; n < 4; ++n) {
                const float2 bp = *(const float2*)&Wt[n * 16 + ml][4 * ks + 2 * g];
                v2f b; b.x = bp.x; b.y = bp.y;
                cacc[0][n] = __builtin_amdgcn_wmma_f32_16x16x4_f32(
                    false, a0, false, b, (short)0, cacc[0][n], false, false);
                cacc[1][n] = __builtin_amdgcn_wmma_f32_16x16x4_f32(
                    false, a1, false, b, (short)0, cacc[1][n], false, false);
            }
        }
    }

    // Scatter: lane holds D[h][v + 8g][n*16 + ml], v = 0..7
#pragma unroll
    for (int h = 0; h < 2; ++h) {
        const int srow = row_base + 16 * h + 8 * g;
#pragma unroll
        for (int v = 0; v < 8; ++v) {
            const int r = srow + v;
            if (r < n_active) {
                const int oi = out_idx[r];
                float* dst = acc + (size_t)oi * COUT + ml;
#pragma unroll
                for (int n = 0; n < 4; ++n)
                    atomicAdd(dst + n * 16, cacc[h][n][v]);
            }
        }
    }
}

// ------------------------------------------------- per-channel stats
__global__ void stats_kernel(const float* __restrict__ acc,
                             const float* __restrict__ bias,
                             float* __restrict__ stats, int total) {
    __shared__ float s[128];
    const int tid = threadIdx.x;
    if (tid < 128) s[tid] = 0.f;
    __syncthreads();
    const int stride = gridDim.x * blockDim.x;
    for (int i = blockIdx.x * blockDim.x + tid; i < total; i += stride) {
        const int ch = i & 63;
        const float v = acc[i] + bias[ch];
        atomicAdd(&s[ch], v);
        atomicAdd(&s[64 + ch], v * v);
    }
    __syncthreads();
    if (tid < 128) atomicAdd(&stats[tid], s[tid]);
}

// ------------------------------------------------- BN + LeakyReLU
__global__ void finalize_kernel(float* __restrict__ out,
                                const float* __restrict__ bias,
                                const float* __restrict__ gamma,
                                const float* __restrict__ beta,
                                const float* __restrict__ stats,
                                int total, float inv_n) {
    const int stride = gridDim.x * blockDim.x;
    for (int i = blockIdx.x * blockDim.x + threadIdx.x; i < total; i += stride) {
        const int ch = i & 63;
        const float v    = out[i] + bias[ch];
        const float mean = stats[ch] * inv_n;
        const float var  = stats[64 + ch] * inv_n - mean * mean;
        const float y    = gamma[ch] * (v - mean) * rsqrtf(var + BN_EPS) + beta[ch];
        out[i] = y > 0.f ? y : LEAK * y;
    }
}

extern "C" void kernel_launch(void* const* d_in, const int* in_sizes, int n_in,
                              void* d_out, int out_size, void* d_ws, size_t ws_size,
                              hipStream_t stream) {
    (void)n_in; (void)ws_size;
    const float* feats    = (const float*)d_in[0];
    const float* weight   = (const float*)d_in[1];
    const float* bias     = (const float*)d_in[2];
    const float* gamma    = (const float*)d_in[3];
    const float* beta     = (const float*)d_in[4];
    const int*   out_idx  = (const int*)d_in[5];
    const int*   kern_idx = (const int*)d_in[6];
    float* out   = (float*)d_out;
    float* stats = (float*)d_ws;    // 128 floats: [sum(64), sumsq(64)]

    const int n_active = in_sizes[0] / CIN;
    const int total    = out_size;             // n_out * 64
    const int n_out    = out_size / COUT;

    {   // zero conv accumulator (d_out) + stats
        int blocks = (total + 255) / 256;
        if (blocks > 4096) blocks = 4096;
        if (blocks < 1) blocks = 1;
        zero_kernel<<<blocks, 256, 0, stream>>>(out, total, stats);
    }
    {   // gather-WMMA-scatter
        const int tiles  = (n_active + ROWS_PER_WAVE - 1) / ROWS_PER_WAVE;
        const int blocks = (tiles + WAVES_PER_BLOCK - 1) / WAVES_PER_BLOCK;
        conv_wmma_kernel<<<blocks, 32 * WAVES_PER_BLOCK, 0, stream>>>(
            feats, weight, out_idx, kern_idx, out, n_active);
    }
    {   // per-channel sum / sumsq
        int blocks = (total + 255) / 256;
        if (blocks > 2048) blocks = 2048;
        if (blocks < 1) blocks = 1;
        stats_kernel<<<blocks, 256, 0, stream>>>(out, bias, stats, total);
    }
    {   // bias + BN + LeakyReLU, in place
        int blocks = (total + 255) / 256;
        if (blocks > 4096) blocks = 4096;
        if (blocks < 1) blocks = 1;
        finalize_kernel<<<blocks, 256, 0, stream>>>(
            out, bias, gamma, beta, stats, total, 1.f / (float)n_out);
    }
}